// TripletOnlyGNN_62130996904304
// MI455X (gfx1250) — compile-verified
//
#include <hip/hip_runtime.h>

// ---------------------------------------------------------------------------
// TripletOnlyGNN on MI455X (gfx1250): fp32 WMMA (v_wmma_f32_16x16x4_f32) GEMMs,
// atomic scatter-mean for the graph aggregations, fused gather for the head.
// ---------------------------------------------------------------------------

typedef __attribute__((ext_vector_type(2))) float v2f;
typedef __attribute__((ext_vector_type(8))) float v8f;

#define N_PEP 20000
#define N_MHC 5000
#define N_TCR 100000
#define EMB   128
#define HID   256
#define E_PM  200000
#define E_MT  400000
#define BATCH 50000

// ---------------------------------------------------------------- utilities

__global__ void zero_f32(float* __restrict__ p, size_t n) {
  size_t i = (size_t)blockIdx.x * blockDim.x + threadIdx.x;
  if (i < n) p[i] = 0.0f;
}

__global__ void edge_count(const int* __restrict__ dst, float* __restrict__ cnt, int E) {
  int e = blockIdx.x * blockDim.x + threadIdx.x;
  if (e < E) atomicAdd(&cnt[dst[e]], 1.0f);
}

template <int SHIFT>
__global__ void edge_scatter(const float* __restrict__ x, const int* __restrict__ src,
                             const int* __restrict__ dst, float* __restrict__ sum,
                             size_t total) {
  size_t i = (size_t)blockIdx.x * blockDim.x + threadIdx.x;
  if (i >= total) return;
  const int D = 1 << SHIFT;
  size_t e = i >> SHIFT;
  int c = (int)(i & (size_t)(D - 1));
  int s = src[e];
  int d = dst[e];
  atomicAdd(&sum[((size_t)d << SHIFT) + c], x[((size_t)s << SHIFT) + c]);
}

template <int SHIFT>
__global__ void mean_div(float* __restrict__ sum, const float* __restrict__ cnt, size_t total) {
  size_t i = (size_t)blockIdx.x * blockDim.x + threadIdx.x;
  if (i >= total) return;
  size_t r = i >> SHIFT;
  sum[i] = sum[i] / fmaxf(cnt[r], 1.0f);
}

// ---------------------------------------------------------------- WMMA GEMM

__device__ __forceinline__ v8f wmma4(v2f a, v2f b, v8f c) {
  // D(16x16,f32) = A(16x4,f32) x B(4x16,f32) + C
  return __builtin_amdgcn_wmma_f32_16x16x4_f32(false, a, false, b, (short)0, c,
                                               false, false);
}

// Accumulate A_row[k0..K) x W[(col..col+48)][k0..K)^T into acc[0..3].
// W is row-major [256 or more rows, ldw cols]; per-lane: r=c=lane&15, g=lane>>4.
__device__ __forceinline__ void mm_panel(const float* __restrict__ arow,
                                         const float* __restrict__ W, int ldw, int K,
                                         int g, int col, v8f acc[4]) {
  for (int k0 = 0; k0 < K; k0 += 4) {
    const int k = k0 + 2 * g;  // lanes 0-15 hold K pair {k0,k0+1}, lanes 16-31 {k0+2,k0+3}
    v2f a = *(const v2f*)(arow + k);
#pragma unroll
    for (int t = 0; t < 4; ++t) {
      v2f b = *(const v2f*)(W + (size_t)(col + 16 * t) * ldw + k);
      acc[t] = wmma4(a, b, acc[t]);
    }
  }
}

// out[M,256] = act( A1 @ W1^T + bias + (A2 ? A2 @ W2^T : 0) ), act = relu?max(0,.)
__global__ __launch_bounds__(128) void sage_gemm(
    const float* __restrict__ A1, const float* __restrict__ W1, int K1,
    const float* __restrict__ A2, const float* __restrict__ W2, int K2,
    const float* __restrict__ bias, float* __restrict__ out, int M, int relu) {
  const int lane = threadIdx.x & 31;
  const int wave = threadIdx.x >> 5;
  const int tile_m = (blockIdx.x * 4 + wave) * 16;
  if (tile_m >= M) return;  // wave-uniform: WMMA below always runs with full EXEC
  const int tile_n = blockIdx.y * 64;
  const int r = lane & 15;
  const int g = lane >> 4;
  const int row = (tile_m + r < M) ? (tile_m + r) : (M - 1);  // clamp for partial tile
  const int col = tile_n + r;

  v8f acc[4] = {v8f{0}, v8f{0}, v8f{0}, v8f{0}};
  mm_panel(A1 + (size_t)row * K1, W1, K1, K1, g, col, acc);
  if (A2) mm_panel(A2 + (size_t)row * K2, W2, K2, K2, g, col, acc);

#pragma unroll
  for (int t = 0; t < 4; ++t) {
    const float bv = bias[col + 16 * t];
#pragma unroll
    for (int j = 0; j < 8; ++j) {
      const int rr = tile_m + j + 8 * g;  // D layout: VGPR j -> row j (lanes 0-15) / j+8
      if (rr < M) {
        float v = acc[t][j] + bv;
        if (relu) v = fmaxf(v, 0.0f);
        out[(size_t)rr * HID + col + 16 * t] = v;
      }
    }
  }
}

// hidden[M,256] = relu( concat(hp[ip], hm[im], ht[it]) @ W^T + bias ), W=[256,768]
__global__ __launch_bounds__(128) void head_gemm(
    const float* __restrict__ hp, const float* __restrict__ hm,
    const float* __restrict__ ht, const int* __restrict__ ip,
    const int* __restrict__ im, const int* __restrict__ it,
    const float* __restrict__ W, const float* __restrict__ bias,
    float* __restrict__ out, int M) {
  const int lane = threadIdx.x & 31;
  const int wave = threadIdx.x >> 5;
  const int tile_m = (blockIdx.x * 4 + wave) * 16;
  if (tile_m >= M) return;
  const int tile_n = blockIdx.y * 64;
  const int r = lane & 15;
  const int g = lane >> 4;
  const int row = (tile_m + r < M) ? (tile_m + r) : (M - 1);
  const int col = tile_n + r;

  const size_t bp = (size_t)ip[row] * HID;
  const size_t bm = (size_t)im[row] * HID;
  const size_t bt = (size_t)it[row] * HID;

  v8f acc[4] = {v8f{0}, v8f{0}, v8f{0}, v8f{0}};
  mm_panel(hp + bp, W,       768, HID, g, col, acc);  // K chunk [0,256)
  mm_panel(hm + bm, W + 256, 768, HID, g, col, acc);  // K chunk [256,512)
  mm_panel(ht + bt, W + 512, 768, HID, g, col, acc);  // K chunk [512,768)

#pragma unroll
  for (int t = 0; t < 4; ++t) {
    const float bv = bias[col + 16 * t];
#pragma unroll
    for (int j = 0; j < 8; ++j) {
      const int rr = tile_m + j + 8 * g;
      if (rr < M) {
        out[(size_t)rr * HID + col + 16 * t] = fmaxf(acc[t][j] + bv, 0.0f);
      }
    }
  }
}

// logits[M] = hidden[M,256] @ w2[256] + b2  (wave-per-row dot, wave32 butterfly)
__global__ __launch_bounds__(256) void logits_kernel(
    const float* __restrict__ hidden, const float* __restrict__ w2,
    const float* __restrict__ b2, float* __restrict__ out, int M) {
  const int lane = threadIdx.x & 31;
  const int wave = threadIdx.x >> 5;
  const int row = blockIdx.x * 8 + wave;
  if (row >= M) return;
  const float* h = hidden + (size_t)row * HID;
  float s = 0.0f;
#pragma unroll
  for (int i = 0; i < HID / 32; ++i) {
    int c = lane + 32 * i;
    s += h[c] * w2[c];
  }
#pragma unroll
  for (int off = 16; off > 0; off >>= 1) s += __shfl_xor(s, off, 32);
  if (lane == 0) out[row] = s + b2[0];
}

// ---------------------------------------------------------------- launcher

extern "C" void kernel_launch(void* const* d_in, const int* in_sizes, int n_in,
                              void* d_out, int out_size, void* d_ws, size_t ws_size,
                              hipStream_t stream) {
  (void)in_sizes; (void)n_in; (void)out_size; (void)ws_size;

  const float* emb_pep = (const float*)d_in[0];
  const float* emb_mhc = (const float*)d_in[1];
  const float* emb_tcr = (const float*)d_in[2];
  const int* src_pm = (const int*)d_in[3];
  const int* dst_pm = (const int*)d_in[4];
  const int* src_mt = (const int*)d_in[5];
  const int* dst_mt = (const int*)d_in[6];
  const int* pack_pep = (const int*)d_in[7];
  const int* pack_mhc = (const int*)d_in[8];
  const int* pack_tcr = (const int*)d_in[9];
  const float* l1_pm_Wl = (const float*)d_in[10];
  const float* l1_pm_bl = (const float*)d_in[11];
  const float* l1_pm_Wr = (const float*)d_in[12];
  const float* l1_mt_Wl = (const float*)d_in[13];
  const float* l1_mt_bl = (const float*)d_in[14];
  const float* l1_mt_Wr = (const float*)d_in[15];
  const float* l2_pm_Wl = (const float*)d_in[16];
  const float* l2_pm_bl = (const float*)d_in[17];
  const float* l2_pm_Wr = (const float*)d_in[18];
  const float* l2_mt_Wl = (const float*)d_in[19];
  const float* l2_mt_bl = (const float*)d_in[20];
  const float* l2_mt_Wr = (const float*)d_in[21];
  const float* proj_W = (const float*)d_in[22];
  const float* proj_b = (const float*)d_in[23];
  const float* head_W1 = (const float*)d_in[24];
  const float* head_b1 = (const float*)d_in[25];
  const float* head_W2 = (const float*)d_in[26];
  const float* head_b2 = (const float*)d_in[27];
  float* logits = (float*)d_out;

  // workspace layout (floats); everything before zero_n is zero-initialized.
  float* ws = (float*)d_ws;
  size_t o = 0;
  float* cnt_pm = ws + o;   o += N_MHC;
  float* cnt_mt = ws + o;   o += N_TCR;
  float* mean_pm = ws + o;  o += (size_t)N_MHC * EMB;   // sum->mean, pep over pm edges
  float* mean_mt1 = ws + o; o += (size_t)N_TCR * EMB;   // sum->mean, mhc over mt edges
  float* mean_mt2 = ws + o; o += (size_t)N_TCR * HID;   // sum->mean, mhc1 over mt edges
  const size_t zero_n = o;
  float* mhc1 = ws + o;  o += (size_t)N_MHC * HID;
  float* mhc2 = ws + o;  o += (size_t)N_MHC * HID;
  float* tcr1 = ws + o;  o += (size_t)N_TCR * HID;
  float* tcr2 = ws + o;  o += (size_t)N_TCR * HID;
  float* hpall = ws + o; o += (size_t)N_PEP * HID;
  float* hidden = mean_mt1;  // dead after tcr1; reuse (both 12.8M floats)

  const int TB = 256;
  auto blks = [](size_t n, int t) { return (unsigned)((n + t - 1) / t); };

  // 1) zero accumulators
  zero_f32<<<blks(zero_n, TB), TB, 0, stream>>>(ws, zero_n);

  // 2) per-destination edge counts (shared by both layers of each relation)
  edge_count<<<blks(E_PM, TB), TB, 0, stream>>>(dst_pm, cnt_pm, E_PM);
  edge_count<<<blks(E_MT, TB), TB, 0, stream>>>(dst_mt, cnt_mt, E_MT);

  // 3) scatter sums + means
  edge_scatter<7><<<blks((size_t)E_PM * EMB, TB), TB, 0, stream>>>(
      emb_pep, src_pm, dst_pm, mean_pm, (size_t)E_PM * EMB);
  edge_scatter<7><<<blks((size_t)E_MT * EMB, TB), TB, 0, stream>>>(
      emb_mhc, src_mt, dst_mt, mean_mt1, (size_t)E_MT * EMB);
  mean_div<7><<<blks((size_t)N_MHC * EMB, TB), TB, 0, stream>>>(
      mean_pm, cnt_pm, (size_t)N_MHC * EMB);
  mean_div<7><<<blks((size_t)N_TCR * EMB, TB), TB, 0, stream>>>(
      mean_mt1, cnt_mt, (size_t)N_TCR * EMB);

  // 4) layer 1 + layer 2 combines (WMMA GEMMs)
  const dim3 blk(128);
  auto ggrid = [](int M) { return dim3((unsigned)((M + 63) / 64), 4); };

  sage_gemm<<<ggrid(N_MHC), blk, 0, stream>>>(mean_pm, l1_pm_Wl, EMB,
                                              emb_mhc, l1_pm_Wr, EMB,
                                              l1_pm_bl, mhc1, N_MHC, 1);
  sage_gemm<<<ggrid(N_TCR), blk, 0, stream>>>(mean_mt1, l1_mt_Wl, EMB,
                                              emb_tcr, l1_mt_Wr, EMB,
                                              l1_mt_bl, tcr1, N_TCR, 1);
  sage_gemm<<<ggrid(N_MHC), blk, 0, stream>>>(mean_pm, l2_pm_Wl, EMB,
                                              mhc1, l2_pm_Wr, HID,
                                              l2_pm_bl, mhc2, N_MHC, 1);

  // 5) layer-2 mt aggregation of mhc1, then tcr2
  edge_scatter<8><<<blks((size_t)E_MT * HID, TB), TB, 0, stream>>>(
      mhc1, src_mt, dst_mt, mean_mt2, (size_t)E_MT * HID);
  mean_div<8><<<blks((size_t)N_TCR * HID, TB), TB, 0, stream>>>(
      mean_mt2, cnt_mt, (size_t)N_TCR * HID);
  sage_gemm<<<ggrid(N_TCR), blk, 0, stream>>>(mean_mt2, l2_mt_Wl, HID,
                                              tcr1, l2_mt_Wr, HID,
                                              l2_mt_bl, tcr2, N_TCR, 1);

  // 6) proj over full pep table (20K rows < 50K gathered rows), no relu
  sage_gemm<<<ggrid(N_PEP), blk, 0, stream>>>(emb_pep, proj_W, EMB,
                                              nullptr, nullptr, 0,
                                              proj_b, hpall, N_PEP, 0);

  // 7) head layer 1 with fused triplet gather (K = 3 x 256)
  head_gemm<<<ggrid(BATCH), blk, 0, stream>>>(hpall, mhc2, tcr2,
                                              pack_pep, pack_mhc, pack_tcr,
                                              head_W1, head_b1, hidden, BATCH);

  // 8) head layer 2 -> logits
  logits_kernel<<<blks(BATCH, 8), 256, 0, stream>>>(hidden, head_W2, head_b2,
                                                    logits, BATCH);
}